// GATConv_30932354465914
// MI455X (gfx1250) — compile-verified
//
#include <hip/hip_runtime.h>

typedef float v2f __attribute__((ext_vector_type(2)));
typedef float v8f __attribute__((ext_vector_type(8)));

typedef unsigned int u32x4 __attribute__((ext_vector_type(4)));
typedef int          i32x4 __attribute__((ext_vector_type(4)));
typedef int          i32x8 __attribute__((ext_vector_type(8)));

#define IN_F 512   // input feature dim
#define HC   128   // H*C output features
#define NH   4     // heads
#define CPH  32    // channels per head

// LDS A-tile row stride: 512 floats + 4 pad floats (2 DWORDs after each 256
// DWORDs, matching the TDM pad_interval/pad_amount scheme). 516 mod 64 = 4
// -> the 16 lanes of a half-wave hit 16 distinct banks (stride 4) on A reads.
#define ROWP (IN_F + 4)
__device__ __forceinline__ int pad_idx(int k) { return k + ((k >> 8) << 1); }

#if defined(__has_builtin)
#if __has_builtin(__builtin_amdgcn_tensor_load_to_lds) && __has_builtin(__builtin_amdgcn_s_wait_tensorcnt)
#define HAVE_TDM 1
#endif
#endif

__device__ __forceinline__ void atomicAddF(float* p, float v) {
    unsafeAtomicAdd(p, v);   // native global_atomic_add_f32 on CDNA5
}

__device__ __forceinline__ float edge_ex(const float* __restrict__ a_s,
                                         const float* __restrict__ a_d,
                                         int s, int d, int h) {
    float x   = a_s[s * NH + h] + a_d[d * NH + h];
    float sig = 1.0f / (1.0f + __expf(-x));
    return __expf(sig);
}

// ---------------------------------------------------------------------------
// Kernel 1: ht = h @ W via V_WMMA_F32_16X16X4_F32.
// Block = 256 threads (8 waves) covering 32 rows x 128 cols.
// A tile (32 x 512 f32, padded rows) staged in LDS by the Tensor Data Mover;
// each wave owns one 16-col strip and keeps TWO accumulators (row halves),
// so every B operand feeds two WMMAs.
// ---------------------------------------------------------------------------
__global__ __launch_bounds__(256) void gat_gemm(const float* __restrict__ h,
                                                const float* __restrict__ W,
                                                float* __restrict__ ht, int N) {
    __shared__ float Atile[32 * ROWP];       // ~64.5 KB of 320 KB/WGP
    const int row0 = blockIdx.x * 32;
    const int t    = threadIdx.x;

#ifdef HAVE_TDM
    if (t < 32) {   // wave 0 only issues the TDM descriptor (uniform branch)
        unsigned long long ga = (unsigned long long)(const void*)(h + (size_t)row0 * IN_F);
        unsigned lds  = (unsigned)(uintptr_t)(&Atile[0]);      // LDS byte offset
        unsigned dim1 = (unsigned)(N - row0);                  // rows left; OOB rows -> 0

        // D# group 0: count=1 | lds_addr | global_addr[56:0] | type=2
        u32x4 g0 = { 1u, lds, (unsigned)ga,
                     (unsigned)((ga >> 32) & 0x01FFFFFFu) | (2u << 30) };
        // D# group 1:
        //  w0: data_size=4B (2<<16) | pad_enable (1<<20) | pad_interval=256dw (7<<22)
        //      | pad_amount=2dw (1<<25)
        //  w1[31:16]=tensor_dim0(512)  w2[31:16]=tensor_dim1.lo
        //  w3[15:0]=tensor_dim1.hi  w3[31:16]=tile_dim0(512)
        //  w4[15:0]=tile_dim1(32)   w5=tensor_dim0_stride(512)
        i32x8 g1 = { (int)((2u << 16) | (1u << 20) | (7u << 22) | (1u << 25)),
                     (int)(512u << 16),
                     (int)((dim1 & 0xFFFFu) << 16),
                     (int)(((dim1 >> 16) & 0xFFFFu) | (512u << 16)),
                     32, 512, 0, 0 };
        i32x4 gz = { 0, 0, 0, 0 };
#if defined(__clang_major__) && (__clang_major__ >= 23)
        i32x8 gz8 = { 0, 0, 0, 0, 0, 0, 0, 0 };
        __builtin_amdgcn_tensor_load_to_lds(g0, g1, gz, gz, gz8, 0);
#else
        __builtin_amdgcn_tensor_load_to_lds(g0, g1, gz, gz, 0);
#endif
        __builtin_amdgcn_s_wait_tensorcnt(0);
    }
#else
    // Fallback: cooperative coalesced load, 64 elements/thread, padded layout.
    for (int i = 0; i < 64; ++i) {
        int idx = t + i * 256;            // 0 .. 16383
        int r   = idx >> 9;               // / 512
        int k   = idx & 511;              // % 512
        int row = row0 + r;
        Atile[r * ROWP + pad_idx(k)] = (row < N) ? h[(size_t)row * IN_F + k] : 0.0f;
    }
#endif
    __syncthreads();

    const int wave = t >> 5;
    const int lane = t & 31;
    const int lr   = lane & 15;           // M index (A) / N index (B,C)
    const int hi   = lane >> 4;           // K-pair select / M-half for C
    const int col  = wave * 16 + lr;

    const float* arow0 = &Atile[lr * ROWP];          // rows 0..15
    const float* arow1 = &Atile[(16 + lr) * ROWP];   // rows 16..31

    v8f acc0 = {};
    v8f acc1 = {};
    for (int k = 0; k < IN_F; k += 4) {
        const int kb  = k + 2 * hi;
        const int kp  = pad_idx(kb);      // kb even -> kb,kb+1 in same pad block
        v2f a0 = { arow0[kp], arow0[kp + 1] };
        v2f a1 = { arow1[kp], arow1[kp + 1] };
        v2f b  = { W[(size_t)kb * HC + col], W[(size_t)(kb + 1) * HC + col] };
        acc0 = __builtin_amdgcn_wmma_f32_16x16x4_f32(false, a0, false, b,
                                                     (short)0, acc0, false, false);
        acc1 = __builtin_amdgcn_wmma_f32_16x16x4_f32(false, a1, false, b,
                                                     (short)0, acc1, false, false);
    }

    // C/D layout: VGPR r -> M = r + 8*hi, N = lr
    if (row0 + 32 <= N) {                 // hot path: no per-row branches
#pragma unroll
        for (int r = 0; r < 8; ++r) {
            ht[(size_t)(row0 + hi * 8 + r) * HC + col]      = acc0[r];
            ht[(size_t)(row0 + 16 + hi * 8 + r) * HC + col] = acc1[r];
        }
    } else {                              // tail block only
        for (int r = 0; r < 8; ++r) {
            int ra = row0 + hi * 8 + r;
            int rb = ra + 16;
            if (ra < N) ht[(size_t)ra * HC + col] = acc0[r];
            if (rb < N) ht[(size_t)rb * HC + col] = acc1[r];
        }
    }
}

// ---------------------------------------------------------------------------
// Kernel 2: per-node, per-head attention logits (wave per node).
// ---------------------------------------------------------------------------
__global__ __launch_bounds__(256) void gat_att(const float* __restrict__ ht,
                                               const float* __restrict__ att_src,
                                               const float* __restrict__ att_dst,
                                               float* __restrict__ a_s,
                                               float* __restrict__ a_d, int N) {
    int node = (int)((blockIdx.x * (size_t)blockDim.x + threadIdx.x) >> 5);
    int lane = threadIdx.x & 31;
    if (node >= N) return;

    float ps[NH], pd[NH];
    for (int i = 0; i < NH; ++i) {
        float v = ht[(size_t)node * HC + i * CPH + lane];
        ps[i] = v * att_src[i * CPH + lane];
        pd[i] = v * att_dst[i * CPH + lane];
    }
    for (int off = 16; off > 0; off >>= 1) {
        for (int i = 0; i < NH; ++i) {
            ps[i] += __shfl_xor(ps[i], off, 32);
            pd[i] += __shfl_xor(pd[i], off, 32);
        }
    }
    if (lane == 0) {
        for (int i = 0; i < NH; ++i) {
            a_s[node * NH + i] = ps[i];
            a_d[node * NH + i] = pd[i];
        }
    }
}

// ---------------------------------------------------------------------------
// Kernel 3: init out with bias broadcast, zero softmax denominators.
// ---------------------------------------------------------------------------
__global__ void gat_init(float* __restrict__ out, float* __restrict__ denom,
                         const float* __restrict__ bias, int N) {
    int t = blockIdx.x * blockDim.x + threadIdx.x;
    if (t < N * HC) out[t] = bias[t & (HC - 1)];
    if (t < N * NH) denom[t] = 0.0f;
}

// ---------------------------------------------------------------------------
// Kernel 4: denom[d,h] += exp(sigmoid(a_s[src]+a_d[dst])) over all edges
// (self-loops: e >= E -> src = dst = e - E).
// ---------------------------------------------------------------------------
__global__ void gat_denom(const int* __restrict__ src, const int* __restrict__ dst,
                          const float* __restrict__ a_s, const float* __restrict__ a_d,
                          float* __restrict__ denom, int E, int N) {
    long long t   = (long long)blockIdx.x * blockDim.x + threadIdx.x;
    long long tot = (long long)(E + N) * NH;
    if (t >= tot) return;
    int e = (int)(t >> 2);
    int h = (int)(t & 3);
    int s, d;
    if (e < E) { s = src[e]; d = dst[e]; } else { s = d = e - E; }
    atomicAddF(&denom[d * NH + h], edge_ex(a_s, a_d, s, d, h));
}

// ---------------------------------------------------------------------------
// Kernel 5: out[d, hc] += ht[s, hc] * alpha(e, h). Thread per (edge, channel):
// coalesced 512B gather of ht[src], coalesced atomic scatter into out[dst].
// ---------------------------------------------------------------------------
__global__ void gat_aggr(const int* __restrict__ src, const int* __restrict__ dst,
                         const float* __restrict__ a_s, const float* __restrict__ a_d,
                         const float* __restrict__ denom, const float* __restrict__ ht,
                         float* __restrict__ out, int E, int N) {
    long long t   = (long long)blockIdx.x * blockDim.x + threadIdx.x;
    long long tot = (long long)(E + N) * HC;
    if (t >= tot) return;
    int e  = (int)(t >> 7);
    int hc = (int)(t & (HC - 1));
    int h  = hc >> 5;
    int s, d;
    if (e < E) { s = src[e]; d = dst[e]; } else { s = d = e - E; }
    float alpha = edge_ex(a_s, a_d, s, d, h) / denom[d * NH + h];
    atomicAddF(&out[(size_t)d * HC + hc], ht[(size_t)s * HC + hc] * alpha);
}

// ---------------------------------------------------------------------------
extern "C" void kernel_launch(void* const* d_in, const int* in_sizes, int n_in,
                              void* d_out, int out_size, void* d_ws, size_t ws_size,
                              hipStream_t stream) {
    const float* h       = (const float*)d_in[0];
    const int*   src     = (const int*)  d_in[1];
    const int*   dst     = (const int*)  d_in[2];
    const float* W       = (const float*)d_in[3];
    const float* att_src = (const float*)d_in[4];
    const float* att_dst = (const float*)d_in[5];
    const float* bias    = (const float*)d_in[6];
    float*       out     = (float*)d_out;

    const int N = in_sizes[0] / IN_F;
    const int E = in_sizes[1];

    // Workspace carve-up (all fp32): ht | a_s | a_d | denom
    float* ws    = (float*)d_ws;
    float* ht    = ws;
    float* a_s   = ht  + (size_t)N * HC;
    float* a_d   = a_s + (size_t)N * NH;
    float* denom = a_d + (size_t)N * NH;

    gat_gemm<<<(N + 31) / 32, 256, 0, stream>>>(h, W, ht, N);
    gat_att<<<(N * 32 + 255) / 256, 256, 0, stream>>>(ht, att_src, att_dst, a_s, a_d, N);
    gat_init<<<(N * HC + 255) / 256, 256, 0, stream>>>(out, denom, bias, N);

    long long tot1 = (long long)(E + N) * NH;
    gat_denom<<<(int)((tot1 + 255) / 256), 256, 0, stream>>>(src, dst, a_s, a_d, denom, E, N);

    long long tot2 = (long long)(E + N) * HC;
    gat_aggr<<<(int)((tot2 + 255) / 256), 256, 0, stream>>>(src, dst, a_s, a_d, denom, ht, out, E, N);
}